// GraphGNN_74560632258710
// MI455X (gfx1250) — compile-verified
//
#include <hip/hip_runtime.h>
#include <math.h>

// Problem constants (from reference)
#define BB   8
#define NN   10000
#define DD   16
#define EE   320000
#define EHID 32
#define EOUT 30
#define NOUT 64

typedef __attribute__((ext_vector_type(2))) float v2f;
typedef __attribute__((ext_vector_type(8))) float v8f;

__device__ __forceinline__ v8f wmma4(v2f a, v2f b, v8f c) {
    // V_WMMA_F32_16X16X4_F32: D(16x16,f32) = A(16x4,f32) x B(4x16,f32) + C
    return __builtin_amdgcn_wmma_f32_16x16x4_f32(
        /*neg_a=*/false, a, /*neg_b=*/false, b,
        /*c_mod=*/(short)0, c, /*reuse_a=*/false, /*reuse_b=*/false);
}

__device__ __forceinline__ float sigmoidf_(float z) {
    return 1.0f / (1.0f + __expf(-z));
}

// ---------------------------------------------------------------- utilities
__global__ void zero_k(float* __restrict__ p, int n) {
    for (int i = blockIdx.x * blockDim.x + threadIdx.x; i < n;
         i += gridDim.x * blockDim.x)
        p[i] = 0.0f;
}

// Pass 1 of edge_attr mean/std: per-block partial {s0,s1,q0,q1}
__global__ __launch_bounds__(256) void ea_partial_k(const float* __restrict__ ea,
                                                    float* __restrict__ part) {
    __shared__ float s[256][4];
    float s0 = 0.f, s1 = 0.f, q0 = 0.f, q1 = 0.f;
    for (int i = blockIdx.x * blockDim.x + threadIdx.x; i < EE;
         i += gridDim.x * blockDim.x) {
        float a = ea[2 * i], b = ea[2 * i + 1];
        s0 += a; s1 += b; q0 += a * a; q1 += b * b;
    }
    int t = threadIdx.x;
    s[t][0] = s0; s[t][1] = s1; s[t][2] = q0; s[t][3] = q1;
    __syncthreads();
    for (int off = 128; off > 0; off >>= 1) {
        if (t < off) {
            s[t][0] += s[t + off][0]; s[t][1] += s[t + off][1];
            s[t][2] += s[t + off][2]; s[t][3] += s[t + off][3];
        }
        __syncthreads();
    }
    if (t == 0) {
        part[blockIdx.x * 4 + 0] = s[0][0];
        part[blockIdx.x * 4 + 1] = s[0][1];
        part[blockIdx.x * 4 + 2] = s[0][2];
        part[blockIdx.x * 4 + 3] = s[0][3];
    }
}

// Pass 2: reduce 256 partials -> {mean0, mean1, 1/std0, 1/std1} (ddof=1)
__global__ __launch_bounds__(256) void ea_stats_k(const float* __restrict__ part,
                                                  float* __restrict__ stats) {
    __shared__ float s[256][4];
    int t = threadIdx.x;
    s[t][0] = part[t * 4 + 0]; s[t][1] = part[t * 4 + 1];
    s[t][2] = part[t * 4 + 2]; s[t][3] = part[t * 4 + 3];
    __syncthreads();
    for (int off = 128; off > 0; off >>= 1) {
        if (t < off) {
            s[t][0] += s[t + off][0]; s[t][1] += s[t + off][1];
            s[t][2] += s[t + off][2]; s[t][3] += s[t + off][3];
        }
        __syncthreads();
    }
    if (t == 0) {
        float m0 = s[0][0] / (float)EE;
        float m1 = s[0][1] / (float)EE;
        float v0 = (s[0][2] - (float)EE * m0 * m0) / (float)(EE - 1);
        float v1 = (s[0][3] - (float)EE * m1 * m1) / (float)(EE - 1);
        stats[0] = m0; stats[1] = m1;
        stats[2] = rsqrtf(v0); stats[3] = rsqrtf(v1);
    }
}

// ----------------------------------------------------- edge MLP + scatter
// One wave handles 16 edges for batch blockIdx.y; block = 8 waves = 128 edges.
__global__ __launch_bounds__(256) void edge_k(
    const float* __restrict__ x, const int* __restrict__ ei,
    const float* __restrict__ ea, const float* __restrict__ wm,
    const float* __restrict__ wsd, const float* __restrict__ W1,
    const float* __restrict__ b1p, const float* __restrict__ W2,
    const float* __restrict__ b2p, const float* __restrict__ stats,
    float* __restrict__ agg)
{
    __shared__ float sW1[36][EHID];               // K-padded 35->36
    __shared__ float sW2[EHID][32];               // N-padded 30->32
    __shared__ __align__(16) float sF[8][16][36]; // feat tile, K-padded
    __shared__ float sH[8][16][EHID];             // hidden tile
    __shared__ int   sSrc[8][16];
    __shared__ int   sTgt[8][16];

    const int tid  = threadIdx.x;
    const int wave = tid >> 5;
    const int lane = tid & 31;
    const int r    = lane & 15;   // row (A/M) or col (B,C/D N) index
    const int h    = lane >> 4;   // lane half
    const int bidx = blockIdx.y;

    // Stage weights (zero-padded)
    for (int i = tid; i < 36 * EHID; i += 256) {
        int k = i / EHID, n = i % EHID;
        sW1[k][n] = (k < 35) ? W1[k * EHID + n] : 0.0f;
    }
    for (int i = tid; i < EHID * 32; i += 256) {
        int k = i >> 5, n = i & 31;
        sW2[k][n] = (n < EOUT) ? W2[k * EOUT + n] : 0.0f;
    }

    // Gather node features + edge scalars into sF (two lanes per edge row)
    const int e   = blockIdx.x * 128 + wave * 16 + r;
    const int src = ei[e];
    const int tgt = ei[EE + e];
    const float4* xs = (const float4*)(x + ((size_t)bidx * NN + src) * DD);
    const float4* xt = (const float4*)(x + ((size_t)bidx * NN + tgt) * DD);
    float4 a0 = xs[2 * h], a1 = xs[2 * h + 1];
    float4 t0 = xt[2 * h], t1 = xt[2 * h + 1];
    *(float4*)&sF[wave][r][8 * h]          = a0;
    *(float4*)&sF[wave][r][8 * h + 4]      = a1;
    *(float4*)&sF[wave][r][16 + 8 * h]     = t0;
    *(float4*)&sF[wave][r][16 + 8 * h + 4] = t1;
    if (h == 0) {
        sSrc[wave][r] = src;
        sTgt[wave][r] = tgt;
    } else {
        // h==1 holds src feats 8..15: a1 = {f12,f13,f14,f15}
        float speed = a1.z * wsd[0] + wm[0];
        float direc = a1.w * wsd[1] + wm[1];
        float cd    = ea[2 * e];
        float cdir  = ea[2 * e + 1];
        float theta = fabsf(cdir - direc);
        float w     = fmaxf(0.0f, 3.0f * speed * cosf(theta) / cd);
        sF[wave][r][32] = (cd   - stats[0]) * stats[2];
        sF[wave][r][33] = (cdir - stats[1]) * stats[3];
        sF[wave][r][34] = w;
        sF[wave][r][35] = 0.0f;
    }
    __syncthreads();

    // GEMM1: H(16x32) = feat(16x36pad) @ W1(36x32), 9 K-steps x 2 N-tiles
    v8f c0 = {}; v8f c1 = {};
    #pragma unroll
    for (int kb = 0; kb < 9; ++kb) {
        int K = 4 * kb + 2 * h;
        v2f a;   a.x   = sF[wave][r][K];   a.y   = sF[wave][r][K + 1];
        v2f bt0; bt0.x = sW1[K][r];        bt0.y = sW1[K + 1][r];
        v2f bt1; bt1.x = sW1[K][16 + r];   bt1.y = sW1[K + 1][16 + r];
        c0 = wmma4(a, bt0, c0);
        c1 = wmma4(a, bt1, c1);
    }
    // bias + sigmoid, restage D-layout -> LDS (for A-layout reload)
    float bh0 = b1p[r], bh1 = b1p[16 + r];
    #pragma unroll
    for (int m = 0; m < 8; ++m) {
        int M = m + 8 * h;
        sH[wave][M][r]      = sigmoidf_(c0[m] + bh0);
        sH[wave][M][16 + r] = sigmoidf_(c1[m] + bh1);
    }
    __syncthreads();

    // GEMM2: E(16x30) = H(16x32) @ W2(32x32pad), 8 K-steps x 2 N-tiles
    v8f d0 = {}; v8f d1 = {};
    #pragma unroll
    for (int kb = 0; kb < 8; ++kb) {
        int K = 4 * kb + 2 * h;
        v2f a;   a.x   = sH[wave][r][K];  a.y   = sH[wave][r][K + 1];
        v2f bt0; bt0.x = sW2[K][r];       bt0.y = sW2[K + 1][r];
        v2f bt1; bt1.x = sW2[K][16 + r];  bt1.y = sW2[K + 1][16 + r];
        d0 = wmma4(a, bt0, d0);
        d1 = wmma4(a, bt1, d1);
    }
    float be0 = b2p[r];
    float be1 = (16 + r < EOUT) ? b2p[16 + r] : 0.0f;
    #pragma unroll
    for (int m = 0; m < 8; ++m) {
        int M  = m + 8 * h;
        int es = sSrc[wave][M];
        int et = sTgt[wave][M];
        float* base_t = agg + ((size_t)bidx * NN + et) * EOUT;
        float* base_s = agg + ((size_t)bidx * NN + es) * EOUT;
        float v0 = sigmoidf_(d0[m] + be0);
        atomicAdd(base_t + r,  v0);
        atomicAdd(base_s + r, -v0);
        if (16 + r < EOUT) {
            float v1 = sigmoidf_(d1[m] + be1);
            atomicAdd(base_t + 16 + r,  v1);
            atomicAdd(base_s + 16 + r, -v1);
        }
    }
}

// --------------------------------------------------------- node MLP
// One wave handles 16 consecutive (b,n) rows; block = 8 waves = 128 rows.
__global__ __launch_bounds__(256) void node_k(const float* __restrict__ agg,
                                              const float* __restrict__ Wn,
                                              const float* __restrict__ bn,
                                              float* __restrict__ out)
{
    __shared__ float sWn[32][NOUT];   // K-padded 30->32
    __shared__ float sA[8][16][32];   // agg tile, K-padded

    const int tid  = threadIdx.x;
    const int wave = tid >> 5;
    const int lane = tid & 31;
    const int r    = lane & 15;
    const int h    = lane >> 4;

    for (int i = tid; i < 32 * NOUT; i += 256) {
        int k = i >> 6, n = i & 63;
        sWn[k][n] = (k < EOUT) ? Wn[k * NOUT + n] : 0.0f;
    }
    const size_t row0 = ((size_t)blockIdx.x * 8 + wave) * 16;
    const float* srcp = agg + row0 * EOUT;
    for (int i = lane; i < 16 * EOUT; i += 32)   // 480 contiguous floats
        sA[wave][i / EOUT][i % EOUT] = srcp[i];
    sA[wave][r][EOUT + h] = 0.0f;                // zero pad cols 30,31
    __syncthreads();

    #pragma unroll
    for (int nt = 0; nt < 4; ++nt) {
        v8f c = {};
        #pragma unroll
        for (int kb = 0; kb < 8; ++kb) {
            int K = 4 * kb + 2 * h;
            v2f a;  a.x  = sA[wave][r][K];        a.y  = sA[wave][r][K + 1];
            v2f bt; bt.x = sWn[K][nt * 16 + r];   bt.y = sWn[K + 1][nt * 16 + r];
            c = wmma4(a, bt, c);
        }
        int col = nt * 16 + r;
        float bias = bn[col];
        #pragma unroll
        for (int m = 0; m < 8; ++m) {
            int M = m + 8 * h;
            out[(row0 + M) * NOUT + col] = sigmoidf_(c[m] + bias);
        }
    }
}

// ----------------------------------------------------------------- launch
extern "C" void kernel_launch(void* const* d_in, const int* in_sizes, int n_in,
                              void* d_out, int out_size, void* d_ws, size_t ws_size,
                              hipStream_t stream) {
    (void)in_sizes; (void)n_in; (void)out_size; (void)ws_size;
    const float* x   = (const float*)d_in[0];
    const int*   ei  = (const int*)d_in[1];
    const float* ea  = (const float*)d_in[2];
    const float* wm  = (const float*)d_in[3];
    const float* wsd = (const float*)d_in[4];
    const float* W1  = (const float*)d_in[5];
    const float* b1  = (const float*)d_in[6];
    const float* W2  = (const float*)d_in[7];
    const float* b2  = (const float*)d_in[8];
    const float* Wn  = (const float*)d_in[9];
    const float* bn  = (const float*)d_in[10];
    float* out = (float*)d_out;

    float* ws    = (float*)d_ws;
    float* stats = ws;           // 4 floats
    float* part  = ws + 4;       // 256*4 floats
    float* agg   = ws + 2048;    // B*N*EOUT floats (9.6 MB), 8KB-offset

    zero_k<<<8, 256, 0, stream>>>(part, 256 * 4);
    zero_k<<<2048, 256, 0, stream>>>(agg, BB * NN * EOUT);
    ea_partial_k<<<256, 256, 0, stream>>>(ea, part);
    ea_stats_k<<<1, 256, 0, stream>>>(part, stats);

    dim3 eg(EE / 128, BB);       // 2500 x 8 blocks, exact tiling
    edge_k<<<eg, 256, 0, stream>>>(x, ei, ea, wm, wsd, W1, b1, W2, b2, stats, agg);

    node_k<<<(BB * NN) / 128, 256, 0, stream>>>(agg, Wn, bn, out);  // 625 blocks
}